// SSIM_33088428048987
// MI455X (gfx1250) — compile-verified
//
#include <hip/hip_runtime.h>
#include <cstdint>
#include <cstddef>

// ---------------------------------------------------------------------------
// Fused SSIM (two-stage 3x3 stencil) for 16x3x512x960 fp32 inputs.
// HBM-bound kernel (~282 MB min traffic -> ~12 us @ 23.3 TB/s).
// CDNA5 path: async global->LDS B128 copies (ASYNCcnt) stage raw x/y tiles;
// both stencil stages run as a register-resident rolling-window sweep
// (no intermediate LDS fields -> ~7x less LDS bank traffic); float4 NT stores.
// ---------------------------------------------------------------------------

typedef float f32x4 __attribute__((ext_vector_type(4)));

namespace {
constexpr int W  = 960;
constexpr int H  = 512;
constexpr int OW = 956;             // output plane: 508 x 956
constexpr int OH = 508;
constexpr int PLANE_IN  = W * H;
constexpr int PLANE_OUT = OW * OH;

constexpr int TIW = 128;            // input tile width (one wave of B128 loads/row)
constexpr int TIH = 76;             // input tile height
constexpr int TOW = 124;            // output tile width  = TIW - 4
constexpr int TOH = 72;             // output tile height = TIH - 4 (8 waves x 9 rows)

constexpr int ROWS_PER_SEG = 9;     // output rows per wave segment
constexpr int INROWS_PER_SEG = ROWS_PER_SEG + 4;  // 13 input rows per lane sweep

constexpr int TILES_X = (OW + TOW - 1) / TOW;   // 8
constexpr int TILES_Y = (OH + TOH - 1) / TOH;   // 8

constexpr float SSIM_C1 = 0.0001f;  // 0.01^2
constexpr float SSIM_C2 = 0.0009f;  // 0.03^2
constexpr float NINTH   = 1.0f / 9.0f;

constexpr size_t SMEM_BYTES = size_t(2) * TIH * TIW * sizeof(float);  // 77,824 B
} // namespace

// Low 32 bits of a generic pointer to LDS == byte offset inside the wave's LDS
// allocation (ISA 10.2) -> what the VDST VGPR of
// global_load_async_to_lds expects.
__device__ __forceinline__ uint32_t lds_byte_off(const void* p) {
  return (uint32_t)(uintptr_t)p;
}

__global__ __launch_bounds__(256)
void ssim_fused_kernel(const float* __restrict__ xg,
                       const float* __restrict__ yg,
                       float* __restrict__ outg) {
  extern __shared__ float smem[];
  float* sX = smem;                 // TIH x TIW raw x tile
  float* sY = sX + TIH * TIW;       // TIH x TIW raw y tile

  const int tid   = threadIdx.x;    // 0..255 (8 wave32s)
  const int lane  = tid & 31;
  const int wave  = tid >> 5;
  const int plane = blockIdx.z;

  int ox = blockIdx.x * TOW; if (ox > OW - TOW) ox = OW - TOW;   // clamped (overlap at edge)
  int oy = blockIdx.y * TOH; if (oy > OH - TOH) oy = OH - TOH;

  // Tile origin in input image == (oy, ox); 16B-aligned (ox multiple of 4).
  const uint64_t xbase =
      (uint64_t)(uintptr_t)(xg + (size_t)plane * PLANE_IN + (size_t)oy * W + ox);
  const uint64_t ybase =
      (uint64_t)(uintptr_t)(yg + (size_t)plane * PLANE_IN + (size_t)oy * W + ox);

  // ---- Async copy: global -> LDS, 16 B per lane, one wave per row ----------
  {
    const uint32_t lx = lds_byte_off(sX);
    const uint32_t ly = lds_byte_off(sY);
#pragma unroll
    for (int it = 0; it < (TIH + 7) / 8; ++it) {     // 10 iterations
      const int r = wave + it * 8;                   // uniform per wave
      if (r < TIH) {
        const uint32_t voff = (uint32_t)((r * W + lane * 4) * 4);    // global byte off
        const uint32_t loff = (uint32_t)((r * TIW + lane * 4) * 4);  // lds byte off
        const uint32_t lax = lx + loff;
        const uint32_t lay = ly + loff;
        asm volatile("global_load_async_to_lds_b128 %0, %1, %2"
                     :: "v"(lax), "v"(voff), "s"(xbase) : "memory");
        asm volatile("global_load_async_to_lds_b128 %0, %1, %2"
                     :: "v"(lay), "v"(voff), "s"(ybase) : "memory");
      }
    }
    asm volatile("s_wait_asynccnt 0" ::: "memory");
  }
  __syncthreads();

  // ---- Register-resident rolling-window sweep ------------------------------
  // wave = row segment (9 output rows), lane (0..30) = 4-output-wide strip.
  if (lane < 31) {
    const int i0 = wave * ROWS_PER_SEG;   // first output row of this segment
    const int j0 = lane * 4;              // first output col of this strip
    float* outp = outg + (size_t)plane * PLANE_OUT
                + (size_t)(oy + i0) * OW + (ox + j0);

    // Ring buffers (fully scalarized after unroll):
    float hx[3][6], hy[3][6];             // horizontal 3-sums of 3 newest input rows
    float cx[2][6], cy[2][6];             // center values of 2 newest input rows
    float a3[3][4], b3[3][4], c3[3][4];   // horiz 3-sums of A,B,C for 3 newest field rows
    float mx[2][4], my[2][4];             // mu centers of 2 newest field rows

#pragma unroll
    for (int rr = 0; rr < INROWS_PER_SEG; ++rr) {    // 13 input rows
      const float* prx = sX + (i0 + rr) * TIW + j0;
      const float* pry = sY + (i0 + rr) * TIW + j0;
      const f32x4 xa = *(const f32x4*)(prx);
      const f32x4 xb = *(const f32x4*)(prx + 4);
      const f32x4 ya = *(const f32x4*)(pry);
      const f32x4 yb = *(const f32x4*)(pry + 4);
      const float xv[8] = {xa.x, xa.y, xa.z, xa.w, xb.x, xb.y, xb.z, xb.w};
      const float yv[8] = {ya.x, ya.y, ya.z, ya.w, yb.x, yb.y, yb.z, yb.w};

      const int kr = rr % 3, kc = rr % 2;
#pragma unroll
      for (int c = 0; c < 6; ++c) {
        hx[kr][c] = xv[c] + xv[c + 1] + xv[c + 2];
        hy[kr][c] = yv[c] + yv[c + 1] + yv[c + 2];
        cx[kc][c] = xv[c + 1];
        cy[kc][c] = yv[c + 1];
      }

      if (rr >= 2) {
        const int m  = rr - 2;            // local field row (center = input row rr-1)
        const int fm = m % 3;
        const int pc = (rr - 1) % 2;      // ring slot holding center row

        float mux[6], muy[6], Af[6], Bf[6], Cf[6];
#pragma unroll
        for (int c = 0; c < 6; ++c) {
          mux[c] = (hx[0][c] + hx[1][c] + hx[2][c]) * NINTH;  // rows m..m+2 (any order)
          muy[c] = (hy[0][c] + hy[1][c] + hy[2][c]) * NINTH;
          const float dx = cx[pc][c] - mux[c];
          const float dy = cy[pc][c] - muy[c];
          Af[c] = dx * dx;
          Bf[c] = dy * dy;
          Cf[c] = dx * dy;
        }
#pragma unroll
        for (int t = 0; t < 4; ++t) {
          a3[fm][t] = Af[t] + Af[t + 1] + Af[t + 2];
          b3[fm][t] = Bf[t] + Bf[t + 1] + Bf[t + 2];
          c3[fm][t] = Cf[t] + Cf[t + 1] + Cf[t + 2];
          mx[m % 2][t] = mux[t + 1];      // field col j+1 for output col j=j0+t
          my[m % 2][t] = muy[t + 1];
        }

        if (m >= 2) {
          const int il = m - 2;           // local output row (uses field rows il..il+2)
          const int ms = (il + 1) % 2;    // mu ring slot of field row il+1
          f32x4 res;
#pragma unroll
          for (int t = 0; t < 4; ++t) {
            const float sigx  = (a3[0][t] + a3[1][t] + a3[2][t]) * NINTH;
            const float sigy  = (b3[0][t] + b3[1][t] + b3[2][t]) * NINTH;
            const float sigxy = (c3[0][t] + c3[1][t] + c3[2][t]) * NINTH;
            const float mu_x = mx[ms][t];
            const float mu_y = my[ms][t];
            const float num = (2.0f * mu_x * mu_y + SSIM_C1) * (2.0f * sigxy + SSIM_C2);
            const float den = (mu_x * mu_x + mu_y * mu_y + SSIM_C1) * (sigx + sigy + SSIM_C2);
            float v = 1.0f - num * __builtin_amdgcn_rcpf(den);   // den >= C1*C2 > 0
            res[t] = v < 0.0f ? 0.0f : (v > 2.0f ? 2.0f : v);
          }
          __builtin_nontemporal_store(res, (f32x4*)(outp + (size_t)il * OW));
        }
      }
    }
  }
}

extern "C" void kernel_launch(void* const* d_in, const int* in_sizes, int n_in,
                              void* d_out, int out_size, void* d_ws, size_t ws_size,
                              hipStream_t stream) {
  (void)n_in; (void)out_size; (void)d_ws; (void)ws_size;
  const float* x = (const float*)d_in[0];
  const float* y = (const float*)d_in[1];
  float* out = (float*)d_out;

  const int planes = in_sizes[0] / PLANE_IN;        // 16*3 = 48
  dim3 grid(TILES_X, TILES_Y, planes);
  dim3 block(256);
  ssim_fused_kernel<<<grid, block, SMEM_BYTES, stream>>>(x, y, out);
}